// _OutputBlock_48455821033949
// MI455X (gfx1250) — compile-verified
//
#include <hip/hip_runtime.h>
#include <hip/hip_bf16.h>

// ---------------------------------------------------------------------------
// Fused edge-MLP + scatter-add for gfx1250 (MI455X), fp32 WMMA path.
//   m:[E,128] -> 3x residual MLP -> silu(linear) -> atomic scatter into [N,128]
// Compute-bound (~115 GFLOP) => V_WMMA_F32_16X16X4_F32, LDS-blocked.
// Work decomposition: 8 waves, each wave = 32 rows x 64 cols so every weight
// (B) fragment loaded from LDS feeds TWO independent WMMA chains.
// ---------------------------------------------------------------------------

#define H      128
#define PITCH  132        // LDS row pitch in floats -> conflict-free fragment loads
#define ROWS   128        // edges per block
#define NWAVES 8          // 256 threads, wave32

typedef float v2f __attribute__((ext_vector_type(2)));
typedef float v8f __attribute__((ext_vector_type(8)));

// silu via v_rcp_f32 (1 ulp) instead of IEEE divide sequence.
__device__ __forceinline__ float silu_f(float x) {
    return x * __builtin_amdgcn_rcpf(1.0f + __expf(-x));
}

// Stage a 128x128 row-major global matrix into LDS with pitch PITCH (float4 moves).
__device__ __forceinline__ void stage_weight(float* __restrict__ WL,
                                             const float* __restrict__ W, int tid) {
    #pragma unroll
    for (int i = 0; i < (H * H / 4) / 256; ++i) {           // 16 iterations
        int idx = tid + i * 256;
        int r  = idx >> 5;                                   // 32 float4 per row
        int c4 = idx & 31;
        float4 v = ((const float4*)W)[idx];                  // idx*4 == r*128 + c4*4
        *(float4*)(WL + r * PITCH + c4 * 4) = v;
    }
}

// Stage the 128-row edge-feature tile (zero padding past E).
__device__ __forceinline__ void stage_m(float* __restrict__ Mb,
                                        const float* __restrict__ m,
                                        long long ebase, long long E, int tid) {
    #pragma unroll
    for (int i = 0; i < (ROWS * H / 4) / 256; ++i) {
        int idx = tid + i * 256;
        int r  = idx >> 5;
        int c4 = idx & 31;
        float4 v = make_float4(0.f, 0.f, 0.f, 0.f);
        long long e = ebase + r;
        if (e < E) v = ((const float4*)(m + e * (long long)H))[c4];
        *(float4*)(Mb + r * PITCH + c4 * 4) = v;
    }
}

// Elementwise pass: dst = silu(src) over a 16-row strip (wave 'w' covers rows
// w*16..w*16+15; all 128 rows covered once per block). Barrier before use.
__device__ __forceinline__ void wave_silu(const float* __restrict__ src,
                                          float* __restrict__ dst,
                                          int rowbase, int lane) {
    #pragma unroll
    for (int i = 0; i < 16; ++i) {                 // 16*128 elems / 32 lanes / 4
        int idx = lane + i * 32;
        int r  = idx >> 5;                          // 32 float4 per row
        int c4 = idx & 31;
        const float* p = src + (rowbase + r) * PITCH + c4 * 4;
        float4 v = *(const float4*)p;
        float4 o;
        o.x = silu_f(v.x); o.y = silu_f(v.y);
        o.z = silu_f(v.z); o.w = silu_f(v.w);
        *(float4*)(dst + (rowbase + r) * PITCH + c4 * 4) = o;
    }
}

#define WMMA_F32(C, A, B) \
    C = __builtin_amdgcn_wmma_f32_16x16x4_f32(false, A, false, B, (short)0, C, \
                                              false, false)

// One H x H linear; this wave computes a 32-row x 64-col output block:
// two 16-row strips (sharing B fragments) x four 16-col tiles.
// MODE 0: OUT[..] = silu(x@W^T + b)        (write ACT)
// MODE 1: MRES[..] += (x@W^T + b)          (residual update of M, in place)
// MODE 2: atomicAdd(gout[node], silu(x@W^T + b))   (final scatter)
template <int MODE>
__device__ __forceinline__ void wave_linear(const float* IN, const float* WL,
                                            const float* __restrict__ bias,
                                            float* OUT, float* MRES,
                                            float* __restrict__ gout,
                                            const int* TGT,
                                            int rbase, int colbase,
                                            int lr, int half) {
    // ---- hoist per-lane bias values (one clause, one wait) ----
    float bv[4];
    #pragma unroll
    for (int t = 0; t < 4; ++t) bv[t] = bias[colbase + t * 16 + lr];

    // ---- hoist scatter targets for MODE 2 ----
    int nodev[16];
    if (MODE == 2) {
        #pragma unroll
        for (int s = 0; s < 2; ++s)
            #pragma unroll
            for (int v = 0; v < 8; ++v)
                nodev[s * 8 + v] = TGT[rbase + s * 16 + v + 8 * half];
    }

    // A(16x4) layout: lane l holds row (l%16), K = 4k + 2*(l/16) + {0,1}.
    const float* in0 = IN + (rbase + lr) * PITCH + 2 * half;        // strip 0
    const float* in1 = in0 + 16 * PITCH;                            // strip 1

    // B(4x16) = W^T: lane l supplies W[col][k], K = 4k + 2*(l/16) + {0,1}.
    const float* w0 = WL + (colbase +  0 + lr) * PITCH + 2 * half;
    const float* w1 = WL + (colbase + 16 + lr) * PITCH + 2 * half;
    const float* w2 = WL + (colbase + 32 + lr) * PITCH + 2 * half;
    const float* w3 = WL + (colbase + 48 + lr) * PITCH + 2 * half;

    // 8 independent accumulation chains: c<strip><tile>
    v8f c00 = {bv[0], bv[0], bv[0], bv[0], bv[0], bv[0], bv[0], bv[0]};
    v8f c01 = {bv[1], bv[1], bv[1], bv[1], bv[1], bv[1], bv[1], bv[1]};
    v8f c02 = {bv[2], bv[2], bv[2], bv[2], bv[2], bv[2], bv[2], bv[2]};
    v8f c03 = {bv[3], bv[3], bv[3], bv[3], bv[3], bv[3], bv[3], bv[3]};
    v8f c10 = c00, c11 = c01, c12 = c02, c13 = c03;

    // K split in halves so only 2x16 A-fragments are live (VGPR <= 256).
    #pragma unroll
    for (int kh = 0; kh < 2; ++kh) {
        v2f a0[16], a1[16];
        #pragma unroll
        for (int k = 0; k < 16; ++k) {
            a0[k] = *(const v2f*)(in0 + 4 * (kh * 16 + k));
            a1[k] = *(const v2f*)(in1 + 4 * (kh * 16 + k));
        }
        #pragma unroll
        for (int k = 0; k < 16; ++k) {
            const int kk = kh * 16 + k;
            v2f f0 = *(const v2f*)(w0 + 4 * kk);
            v2f f1 = *(const v2f*)(w1 + 4 * kk);
            v2f f2 = *(const v2f*)(w2 + 4 * kk);
            v2f f3 = *(const v2f*)(w3 + 4 * kk);
            // each B fragment feeds both row strips (2 WMMAs per ds_load_b64)
            WMMA_F32(c00, a0[k], f0);  WMMA_F32(c10, a1[k], f0);
            WMMA_F32(c01, a0[k], f1);  WMMA_F32(c11, a1[k], f1);
            WMMA_F32(c02, a0[k], f2);  WMMA_F32(c12, a1[k], f2);
            WMMA_F32(c03, a0[k], f3);  WMMA_F32(c13, a1[k], f3);
        }
    }

    // ---- epilogue: C element v lives at (row = v + 8*half, col = lr) ----
    #pragma unroll
    for (int v = 0; v < 8; ++v) {
        const int r0 = rbase + v + 8 * half;          // strip 0 row
        const int r1 = r0 + 16;                       // strip 1 row
        const int k0 = colbase +  0 + lr;
        const int k1 = colbase + 16 + lr;
        const int k2 = colbase + 32 + lr;
        const int k3 = colbase + 48 + lr;
        if (MODE == 0) {
            OUT[r0 * PITCH + k0] = silu_f(c00[v]);
            OUT[r0 * PITCH + k1] = silu_f(c01[v]);
            OUT[r0 * PITCH + k2] = silu_f(c02[v]);
            OUT[r0 * PITCH + k3] = silu_f(c03[v]);
            OUT[r1 * PITCH + k0] = silu_f(c10[v]);
            OUT[r1 * PITCH + k1] = silu_f(c11[v]);
            OUT[r1 * PITCH + k2] = silu_f(c12[v]);
            OUT[r1 * PITCH + k3] = silu_f(c13[v]);
        } else if (MODE == 1) {
            MRES[r0 * PITCH + k0] += c00[v];
            MRES[r0 * PITCH + k1] += c01[v];
            MRES[r0 * PITCH + k2] += c02[v];
            MRES[r0 * PITCH + k3] += c03[v];
            MRES[r1 * PITCH + k0] += c10[v];
            MRES[r1 * PITCH + k1] += c11[v];
            MRES[r1 * PITCH + k2] += c12[v];
            MRES[r1 * PITCH + k3] += c13[v];
        } else {
            const int n0 = nodev[v];
            const int n1 = nodev[8 + v];
            if (n0 >= 0) {
                atomicAdd(gout + (long long)n0 * H + k0, silu_f(c00[v]));
                atomicAdd(gout + (long long)n0 * H + k1, silu_f(c01[v]));
                atomicAdd(gout + (long long)n0 * H + k2, silu_f(c02[v]));
                atomicAdd(gout + (long long)n0 * H + k3, silu_f(c03[v]));
            }
            if (n1 >= 0) {
                atomicAdd(gout + (long long)n1 * H + k0, silu_f(c10[v]));
                atomicAdd(gout + (long long)n1 * H + k1, silu_f(c11[v]));
                atomicAdd(gout + (long long)n1 * H + k2, silu_f(c12[v]));
                atomicAdd(gout + (long long)n1 * H + k3, silu_f(c13[v]));
            }
        }
    }
}

__global__ void __launch_bounds__(256)
edge_mlp_scatter_kernel(const float* __restrict__ m, const int* __restrict__ tgt,
                        const float* __restrict__ W1, const float* __restrict__ b1,
                        const float* __restrict__ W2, const float* __restrict__ b2,
                        const float* __restrict__ Wlin, const float* __restrict__ blin,
                        float* __restrict__ out, long long E, int L) {
    extern __shared__ float smem[];
    float* Mb  = smem;                       // ROWS x PITCH  (raw residual state)
    float* SIN = Mb + ROWS * PITCH;          // ROWS x PITCH  (silu(M), matmul-1 input)
    float* ACT = SIN + ROWS * PITCH;         // ROWS x PITCH  (silu(h), matmul-2 input)
    float* WL  = ACT + ROWS * PITCH;         // H x PITCH     (staged weight matrix)
    int*   TGT = (int*)(WL + H * PITCH);     // ROWS node ids

    const int tid = threadIdx.x;
    const long long ebase = (long long)blockIdx.x * ROWS;

    stage_m(Mb, m, ebase, E, tid);
    if (tid < ROWS) {
        long long e = ebase + tid;
        TGT[tid] = (e < E) ? tgt[e] : -1;
    }

    const int lane    = tid & 31;
    const int wave    = tid >> 5;
    const int silubase = wave * 16;          // elementwise-pass partition
    const int rbase   = (wave & 3) * 32;     // matmul row-strip pair
    const int colbase = (wave >> 2) * 64;    // matmul column half
    const int lr      = lane & 15;
    const int half    = lane >> 4;

    for (int l = 0; l < L; ++l) {
        __syncthreads();                                   // Mb/ACT ready, WL free
        stage_weight(WL, W1 + (long long)l * H * H, tid);
        wave_silu(Mb, SIN, silubase, lane);                // hides staging latency
        __syncthreads();
        // ACT = silu( SIN @ W1^T + b1 )
        wave_linear<0>(SIN, WL, b1 + l * H, ACT, nullptr, nullptr, nullptr,
                       rbase, colbase, lr, half);
        __syncthreads();
        stage_weight(WL, W2 + (long long)l * H * H, tid);
        __syncthreads();
        // M += ACT @ W2^T + b2
        wave_linear<1>(ACT, WL, b2 + l * H, nullptr, Mb, nullptr, nullptr,
                       rbase, colbase, lr, half);
    }

    __syncthreads();
    stage_weight(WL, Wlin, tid);
    wave_silu(Mb, SIN, silubase, lane);
    __syncthreads();
    // out[node] += silu( SIN @ Wlin^T + blin )
    wave_linear<2>(SIN, WL, blin, nullptr, nullptr, out, TGT,
                   rbase, colbase, lr, half);
}

__global__ void zero_f32_kernel(float* __restrict__ p, long long n) {
    long long i = (long long)blockIdx.x * blockDim.x + threadIdx.x;
    if (i < n) p[i] = 0.0f;
}

extern "C" void kernel_launch(void* const* d_in, const int* in_sizes, int n_in,
                              void* d_out, int out_size, void* d_ws, size_t ws_size,
                              hipStream_t stream) {
    const float* m          = (const float*)d_in[0];
    const int*   edge_index = (const int*)d_in[1];   // [2, E], targets are row 1
    const float* W1         = (const float*)d_in[3];
    const float* b1         = (const float*)d_in[4];
    const float* W2         = (const float*)d_in[5];
    const float* b2         = (const float*)d_in[6];
    const float* Wlin       = (const float*)d_in[7];
    const float* blin       = (const float*)d_in[8];
    float* out = (float*)d_out;

    const long long E = in_sizes[0] / H;
    const int       L = in_sizes[3] / (H * H);
    const int* tgt = edge_index + E;                 // edge_index[1]

    // Kernel accumulates into out -> zero it every launch (graph-replay safe).
    {
        long long n = out_size;
        int blocks = (int)((n + 1023) / 1024);
        zero_f32_kernel<<<blocks, 1024, 0, stream>>>(out, n);
    }

    const size_t shmem = (size_t)(3 * ROWS * PITCH + H * PITCH) * sizeof(float)
                       + (size_t)ROWS * sizeof(int);   // ~265 KB (<= 320 KB/WGP)
    (void)hipFuncSetAttribute((const void*)edge_mlp_scatter_kernel,
                              hipFuncAttributeMaxDynamicSharedMemorySize,
                              (int)shmem);

    const int blocks = (int)((E + ROWS - 1) / ROWS);
    edge_mlp_scatter_kernel<<<blocks, 256, shmem, stream>>>(
        m, tgt, W1, b1, W2, b2, Wlin, blin, out, E, L);
}